// HGCN_22136261444116
// MI455X (gfx1250) — compile-verified
//
#include <hip/hip_runtime.h>
#include <hip/hip_bf16.h>

#define MIN_NORM   1e-15f
#define PROJ_EPS   4e-3f
#define ATANH_CLIP (1.0f - 1e-7f)

typedef __attribute__((ext_vector_type(2))) float v2f;
typedef __attribute__((ext_vector_type(8))) float v8f;

// ---------- wave32 helpers ----------
__device__ __forceinline__ float wred(float v) {
    v += __shfl_xor(v, 16, 32);
    v += __shfl_xor(v, 8, 32);
    v += __shfl_xor(v, 4, 32);
    v += __shfl_xor(v, 2, 32);
    v += __shfl_xor(v, 1, 32);
    return v;
}
__device__ __forceinline__ float dot4(float4 a, float4 b) {
    return a.x * b.x + a.y * b.y + a.z * b.z + a.w * b.w;
}
__device__ __forceinline__ float4 f4scale(float4 a, float s) {
    return make_float4(a.x * s, a.y * s, a.z * s, a.w * s);
}
// ||row|| with MIN_NORM clamp (row = 32 lanes x float4)
__device__ __forceinline__ float wnorm(float4 v) {
    return fmaxf(sqrtf(wred(dot4(v, v))), MIN_NORM);
}
__device__ __forceinline__ float artanh_c(float x) {
    x = fminf(fmaxf(x, -ATANH_CLIP), ATANH_CLIP);
    return 0.5f * (log1pf(x) - log1pf(-x));
}
// scale factor implementing _proj (clamp to Poincare ball of curvature c, sc = sqrt(c))
__device__ __forceinline__ float proj_scale(float n, float sc) {
    float maxn = (1.0f - PROJ_EPS) / sc;
    return (n > maxn) ? (maxn / n) : 1.0f;
}

// ---------- proj(expmap0(u, c)) row kernel (also used for the bias vector) ----------
__global__ __launch_bounds__(256) void k_expmap_proj(const float* __restrict__ in,
                                                     float* __restrict__ out,
                                                     float c, int nrows) {
    int row  = (int)((blockIdx.x * blockDim.x + threadIdx.x) >> 5);
    int lane = threadIdx.x & 31;
    if (row >= nrows) return;
    float  sc = sqrtf(c);
    size_t off = (size_t)row * 128 + lane * 4;
    float4 u  = *(const float4*)(in + off);
    float  un = wnorm(u);
    float4 h  = f4scale(u, tanhf(sc * un) / (sc * un));
    float  hn = wnorm(h);
    h = f4scale(h, proj_scale(hn, sc));
    *(float4*)(out + off) = h;
}

// ---------- pack W[128][128] into WMMA B-fragment order ----------
// Entry i = (ct*16 + kb2)*32 + lane holds the 4 B-matrix values lane needs for
// k-steps 2*kb2 and 2*kb2+1 of column tile ct:
//   j  = ct*16 + (lane&15)        (N index)
//   k0 = kb2*8 + 2*(lane>>4)      (K index; B[k][j] = W[j][k])
//   {W[j][k0], W[j][k0+1], W[j][k0+4], W[j][k0+5]}
// In the GEMM this makes B loads one coalesced global_load_b128 per 2 WMMAs.
__global__ __launch_bounds__(256) void k_pack_w(const float* __restrict__ W,
                                                float* __restrict__ P) {
    int i = blockIdx.x * blockDim.x + threadIdx.x;
    if (i >= 8 * 16 * 32) return;
    int ct   = i >> 9;
    int rem  = i & 511;
    int kb2  = rem >> 5;
    int lane = rem & 31;
    int j  = ct * 16 + (lane & 15);
    int k0 = kb2 * 8 + (lane >> 4) * 2;
    float4 v = make_float4(W[j * 128 + k0],     W[j * 128 + k0 + 1],
                           W[j * 128 + k0 + 4], W[j * 128 + k0 + 5]);
    *(float4*)(P + (size_t)i * 4) = v;
}

// ---------- fp32 WMMA GEMM: Out[N,128] = H[N,128] @ W^T (P = packed W frags) ----------
// Block = 256 threads = 8 waves; block owns 16 rows; wave w owns columns [16w,16w+16).
__global__ __launch_bounds__(256) void k_gemm_wmma(const float* __restrict__ H,
                                                   const float* __restrict__ P,
                                                   float* __restrict__ Out, int nrows) {
    __shared__ float As[16 * 132];  // stride 132 floats: 16B-aligned, no bank conflicts
    int rowBase = blockIdx.x * 16;
    int tid  = threadIdx.x;
    int wave = tid >> 5;
    int lane = tid & 31;

    // cooperative load: 16 rows x 128 floats, 8 floats per thread
#pragma unroll
    for (int i = 0; i < 2; ++i) {
        int idx = tid * 8 + i * 4;
        int r = idx >> 7, c = idx & 127;
        int g = rowBase + r;
        float4 v = make_float4(0.f, 0.f, 0.f, 0.f);
        if (g < nrows) v = *(const float4*)(H + (size_t)g * 128 + c);
        *(float4*)(&As[r * 132 + c]) = v;
    }
    __syncthreads();

    int colBase = wave * 16;
    int mrow = lane & 15;         // A: M index / B,C,D: N index
    int koff = (lane >> 4) * 2;   // lanes 16-31 hold K+2,K+3
    const float4* Pw = ((const float4*)P) + (size_t)wave * 16 * 32 + lane;

    v8f acc = {};
#pragma unroll
    for (int kb2 = 0; kb2 < 16; ++kb2) {
        float4 bp = Pw[kb2 * 32];                 // one b128: B frags for 2 k-steps
        int k0 = kb2 * 8 + koff;
        v2f a0, a1, b0, b1;
        a0[0] = As[mrow * 132 + k0];     a0[1] = As[mrow * 132 + k0 + 1];
        a1[0] = As[mrow * 132 + k0 + 4]; a1[1] = As[mrow * 132 + k0 + 5];
        b0[0] = bp.x; b0[1] = bp.y;
        b1[0] = bp.z; b1[1] = bp.w;
        acc = __builtin_amdgcn_wmma_f32_16x16x4_f32(false, a0, false, b0,
                                                    (short)0, acc, false, false);
        acc = __builtin_amdgcn_wmma_f32_16x16x4_f32(false, a1, false, b1,
                                                    (short)0, acc, false, false);
    }
    // C/D layout: VGPR v, lane L -> M = v + 8*L[4], N = L[3:0]
#pragma unroll
    for (int v = 0; v < 8; ++v) {
        int r = rowBase + v + (lane >> 4) * 8;
        if (r < nrows) Out[(size_t)r * 128 + colBase + mrow] = acc[v];
    }
}

// ---------- mobius_matvec tail + proj + mobius_add(bias) + proj + logmap0 ----------
// Reads H (layer input) and MX (GEMM result), writes ht in-place over MX.
__global__ __launch_bounds__(256) void k_rowops1(const float* __restrict__ H,
                                                 float* __restrict__ MX,
                                                 const float* __restrict__ hb,
                                                 float c, int nrows) {
    int row  = (int)((blockIdx.x * blockDim.x + threadIdx.x) >> 5);
    int lane = threadIdx.x & 31;
    if (row >= nrows) return;
    float  sc  = sqrtf(c);
    size_t off = (size_t)row * 128 + lane * 4;
    float4 x  = *(const float4*)(H + off);
    float4 m  = *(const float4*)(MX + off);
    float4 bv = *(const float4*)(hb + lane * 4);

    float xn  = wnorm(x);
    float mxn = wnorm(m);
    float s   = tanhf((mxn / xn) * artanh_c(sc * xn)) / (mxn * sc);
    float4 res = f4scale(m, s);
    int lz = (m.x == 0.f) && (m.y == 0.f) && (m.z == 0.f) && (m.w == 0.f);
    if (__all(lz)) res = make_float4(0.f, 0.f, 0.f, 0.f);
    // proj
    res = f4scale(res, proj_scale(wnorm(res), sc));
    // mobius_add(res, hyp_b)
    float x2 = wred(dot4(res, res));
    float y2 = wred(dot4(bv, bv));
    float xy = wred(dot4(res, bv));
    float ka  = 1.0f + 2.0f * c * xy + c * y2;
    float kb  = 1.0f - c * x2;
    float den = fmaxf(1.0f + 2.0f * c * xy + c * c * x2 * y2, MIN_NORM);
    float4 h2 = make_float4((ka * res.x + kb * bv.x) / den,
                            (ka * res.y + kb * bv.y) / den,
                            (ka * res.z + kb * bv.z) / den,
                            (ka * res.w + kb * bv.w) / den);
    // proj
    h2 = f4scale(h2, proj_scale(wnorm(h2), sc));
    // logmap0
    float pn = wnorm(h2);
    float ls = artanh_c(sc * pn) / (sc * pn);
    *(float4*)(MX + off) = f4scale(h2, ls);
}

// ---------- zero the aggregation buffer ----------
__global__ __launch_bounds__(256) void k_zero(float* __restrict__ p, size_t n4) {
    size_t i = (size_t)blockIdx.x * blockDim.x + threadIdx.x;
    if (i < n4) ((float4*)p)[i] = make_float4(0.f, 0.f, 0.f, 0.f);
}

// ---------- edge scatter: agg[dst] += ht[src] * w (one wave per edge) ----------
__global__ __launch_bounds__(256) void k_scatter(const float* __restrict__ ht,
                                                 const long long* __restrict__ src,
                                                 const long long* __restrict__ dst,
                                                 const float* __restrict__ ew,
                                                 float* __restrict__ agg, int nedges) {
    int e    = (int)((blockIdx.x * blockDim.x + threadIdx.x) >> 5);
    int lane = threadIdx.x & 31;
    if (e >= nedges) return;
    long long s = src[e], d = dst[e];
    float w = ew[e];
    float4 v = *(const float4*)(ht + (size_t)s * 128 + lane * 4);
    float* p = agg + (size_t)d * 128 + lane * 4;
    unsafeAtomicAdd(p + 0, v.x * w);   // -> global_atomic_add_f32
    unsafeAtomicAdd(p + 1, v.y * w);
    unsafeAtomicAdd(p + 2, v.z * w);
    unsafeAtomicAdd(p + 3, v.w * w);
}

// ---------- proj(expmap0(.,c_in)) -> logmap0 -> relu -> proj(expmap0(.,c_out)) ----------
__global__ __launch_bounds__(256) void k_rowops2(const float* __restrict__ AGG,
                                                 float* __restrict__ OUT,
                                                 float c_in, float c_out, int nrows) {
    int row  = (int)((blockIdx.x * blockDim.x + threadIdx.x) >> 5);
    int lane = threadIdx.x & 31;
    if (row >= nrows) return;
    float sci = sqrtf(c_in), sco = sqrtf(c_out);
    size_t off = (size_t)row * 128 + lane * 4;
    float4 a  = *(const float4*)(AGG + off);
    float  an = wnorm(a);
    float4 h  = f4scale(a, tanhf(sci * an) / (sci * an));
    h = f4scale(h, proj_scale(wnorm(h), sci));
    float pn = wnorm(h);
    float4 xt = f4scale(h, artanh_c(sci * pn) / (sci * pn));
    xt.x = fmaxf(xt.x, 0.f); xt.y = fmaxf(xt.y, 0.f);
    xt.z = fmaxf(xt.z, 0.f); xt.w = fmaxf(xt.w, 0.f);
    float  un = wnorm(xt);
    float4 o  = f4scale(xt, tanhf(sco * un) / (sco * un));
    o = f4scale(o, proj_scale(wnorm(o), sco));
    *(float4*)(OUT + off) = o;
}

extern "C" void kernel_launch(void* const* d_in, const int* in_sizes, int n_in,
                              void* d_out, int out_size, void* d_ws, size_t ws_size,
                              hipStream_t stream) {
    const float*     x  = (const float*)d_in[0];
    const float*     W1 = (const float*)d_in[1];
    const float*     b1 = (const float*)d_in[2];
    const float*     W2 = (const float*)d_in[3];
    const float*     b2 = (const float*)d_in[4];
    const long long* ei = (const long long*)d_in[5];   // int64 [2, E]
    const float*     ew = (const float*)d_in[6];

    int nrows  = in_sizes[0] / 128;
    int nedges = in_sizes[6];
    const long long* src = ei;
    const long long* dst = ei + nedges;

    float* bufH = (float*)d_ws;                         // layer input h
    float* bufM = bufH + (size_t)nrows * 128;           // GEMM out -> ht (in place)
    float* bufA = bufM + (size_t)nrows * 128;           // aggregation
    float* wp1  = bufA + (size_t)nrows * 128;           // packed W1 fragments
    float* wp2  = wp1 + 128 * 128;                      // packed W2 fragments
    float* hb1  = wp2 + 128 * 128;                      // proj(expmap0(b1, c))
    float* hb2  = hb1 + 128;

    const float C_IN = 1.0f, C_HID = 1.25f, C_OUT = 1.5f;
    dim3  blk(256);
    int   rowBlocks  = (nrows + 7) / 8;
    int   gemmBlocks = (nrows + 15) / 16;
    size_t n4        = (size_t)nrows * 128 / 4;
    int   zeroBlocks = (int)((n4 + 255) / 256);
    int   edgeBlocks = (nedges + 7) / 8;

    k_pack_w<<<16, 256, 0, stream>>>(W1, wp1);
    k_pack_w<<<16, 256, 0, stream>>>(W2, wp2);
    k_expmap_proj<<<1, 32, 0, stream>>>(b1, hb1, C_IN, 1);
    k_expmap_proj<<<1, 32, 0, stream>>>(b2, hb2, C_HID, 1);
    k_expmap_proj<<<rowBlocks, blk, 0, stream>>>(x, bufH, C_IN, nrows);

    // ----- layer 1: c_in = 1.0 -> c_out = 1.25 -----
    k_gemm_wmma<<<gemmBlocks, blk, 0, stream>>>(bufH, wp1, bufM, nrows);
    k_rowops1<<<rowBlocks, blk, 0, stream>>>(bufH, bufM, hb1, C_IN, nrows);
    k_zero<<<zeroBlocks, blk, 0, stream>>>(bufA, n4);
    k_scatter<<<edgeBlocks, blk, 0, stream>>>(bufM, src, dst, ew, bufA, nedges);
    k_rowops2<<<rowBlocks, blk, 0, stream>>>(bufA, bufH, C_IN, C_HID, nrows);

    // ----- layer 2: c_in = 1.25 -> c_out = 1.5 -----
    k_gemm_wmma<<<gemmBlocks, blk, 0, stream>>>(bufH, wp2, bufM, nrows);
    k_rowops1<<<rowBlocks, blk, 0, stream>>>(bufH, bufM, hb2, C_HID, nrows);
    k_zero<<<zeroBlocks, blk, 0, stream>>>(bufA, n4);
    k_scatter<<<edgeBlocks, blk, 0, stream>>>(bufM, src, dst, ew, bufA, nedges);
    k_rowops2<<<rowBlocks, blk, 0, stream>>>(bufA, (float*)d_out, C_HID, C_OUT, nrows);
}